// RoIHeads_29918742184608
// MI455X (gfx1250) — compile-verified
//
#include <hip/hip_runtime.h>
#include <math.h>

// Problem constants (match reference)
static constexpr int   Bq    = 4;
static constexpr int   Nq    = 4096;
static constexpr int   Cq    = 91;
static constexpr int   CM1   = Cq - 1;        // 90 foreground classes
static constexpr int   Mq    = Nq * CM1;      // 368640 candidates / image
static constexpr int   DETS  = 100;
static constexpr float CLIPV   = 4.135166556742356f;  // log(1000/16)
static constexpr float SCORE_T = 0.05f;
static constexpr float NMS_T   = 0.5f;
static constexpr float MINSZ   = 0.01f;

typedef __attribute__((ext_vector_type(2))) float v2f;
typedef __attribute__((ext_vector_type(8))) float v8f;

// ---------------------------------------------------------------------------
// Kernel 1: fused softmax + box decode + filter.
// One wave32 handles 16 rows. Each lane owns 2 of every 4 columns of its row
// (A-matrix layout of V_WMMA_F32_16X16X4_F32: lane L = (h=L>>4, r=L&15)
// supplies A[r][k0+2h], A[r][k0+2h+1]). All 46 owned logits are loaded into
// registers once; exp() is computed once and reused by both the WMMA row-sum
// and the decode pass. Two accumulators break the WMMA RAW chain.
// ---------------------------------------------------------------------------
__global__ __launch_bounds__(256)
void prep_kernel(const float* __restrict__ logits,
                 const float* __restrict__ deltas,
                 const float* __restrict__ props,
                 const int*   __restrict__ shapes,
                 float4* __restrict__ boxes_ws,
                 float*  __restrict__ live_ws)
{
    const int gtid = blockIdx.x * blockDim.x + threadIdx.x;
    const int wid  = gtid >> 5;
    const int lane = gtid & 31;
    const int h    = lane >> 4;      // which half of the wave
    const int r    = lane & 15;      // row within the 16-row tile
    const int row  = wid * 16 + r;   // global row in [0, B*N)
    const int img  = row / Nq;
    const int n    = row % Nq;

    // proposal (xyxy) for this row
    const float4 p  = ((const float4*)props)[row];
    const float pw  = p.z - p.x;
    const float ph2 = p.w - p.y;
    const float pcx = p.x + 0.5f * pw;
    const float pcy = p.y + 0.5f * ph2;
    const float Hh  = (float)shapes[img * 2 + 0];
    const float Ww  = (float)shapes[img * 2 + 1];

    const float* lrow = logits + (size_t)row * Cq;

    // ---- load all owned logits once (addresses clamped in-bounds) ----
    float lg[46];
#pragma unroll
    for (int ch = 0; ch < 23; ++ch) {
        const int c0 = ch * 4 + h * 2;            // <= 90 always
        const int c1 = (c0 + 1 < Cq) ? (c0 + 1) : (Cq - 1);
        lg[2 * ch]     = lrow[c0];
        lg[2 * ch + 1] = lrow[c1];
    }
    // only invalid slot: ch=22, h=1, second element (column 91)
    if (h == 1) lg[45] = -INFINITY;

    // ---- row max over registers, combine with partner half ----
    float mx = -INFINITY;
#pragma unroll
    for (int i = 0; i < 46; ++i) mx = fmaxf(mx, lg[i]);
    mx = fmaxf(mx, __shfl_xor(mx, 16));

    // ---- exp in place (exp(-inf)=0 masks the invalid slot for the sum) ----
#pragma unroll
    for (int i = 0; i < 46; ++i) lg[i] = __expf(lg[i] - mx);

    // ---- WMMA row sums: D += A(16x4 exps) x B(ones), two-deep pipeline ----
    v8f acc0 = {}, acc1 = {};
    const v2f bones = {1.0f, 1.0f};
#pragma unroll
    for (int ch = 0; ch < 23; ch += 2) {
        v2f a; a.x = lg[2 * ch]; a.y = lg[2 * ch + 1];
        acc0 = __builtin_amdgcn_wmma_f32_16x16x4_f32(
                   false, a, false, bones, (short)0, acc0, false, false);
        if (ch + 1 < 23) {
            v2f b2; b2.x = lg[2 * ch + 2]; b2.y = lg[2 * ch + 3];
            acc1 = __builtin_amdgcn_wmma_f32_16x16x4_f32(
                       false, b2, false, bones, (short)0, acc1, false, false);
        }
    }

    // D layout: VGPR j holds D[8h + j][col]; row r lives in VGPR (r&7), half (r>>3)
    const int j7 = r & 7;
    const float m0 =
        (j7 & 4) ? ((j7 & 2) ? ((j7 & 1) ? acc0[7] : acc0[6])
                             : ((j7 & 1) ? acc0[5] : acc0[4]))
                 : ((j7 & 2) ? ((j7 & 1) ? acc0[3] : acc0[2])
                             : ((j7 & 1) ? acc0[1] : acc0[0]));
    const float m1 =
        (j7 & 4) ? ((j7 & 2) ? ((j7 & 1) ? acc1[7] : acc1[6])
                             : ((j7 & 1) ? acc1[5] : acc1[4]))
                 : ((j7 & 2) ? ((j7 & 1) ? acc1[3] : acc1[2])
                             : ((j7 & 1) ? acc1[1] : acc1[0]));
    const float mine  = m0 + m1;
    const float other = __shfl_xor(mine, 16);
    const float rsum  = ((r >> 3) == h) ? mine : other;
    const float inv   = 1.0f / rsum;

    // ---- decode + clip + filter + emit candidates (scores from registers) ----
    const float* drow = deltas + (size_t)row * (Cq * 4);
    float4* bx = boxes_ws + (size_t)img * Mq;
    float*  lv = live_ws  + (size_t)img * Mq;

#pragma unroll
    for (int ch = 0; ch < 23; ++ch) {
#pragma unroll
        for (int t = 0; t < 2; ++t) {
            const int c = ch * 4 + h * 2 + t;
            if (c >= 1 && c < Cq) {          // skip background class 0
                const float sc = lg[2 * ch + t] * inv;
                const float4 dl = *(const float4*)(drow + 4 * c); // 16B aligned
                const float dx = dl.x * 0.1f;
                const float dy = dl.y * 0.1f;
                const float dw = fminf(dl.z * 0.2f, CLIPV);
                const float dh = fminf(dl.w * 0.2f, CLIPV);
                const float qcx = dx * pw  + pcx;
                const float qcy = dy * ph2 + pcy;
                const float qw  = __expf(dw) * pw;
                const float qh  = __expf(dh) * ph2;
                const float x1 = fminf(fmaxf(qcx - 0.5f * qw, 0.0f), Ww);
                const float y1 = fminf(fmaxf(qcy - 0.5f * qh, 0.0f), Hh);
                const float x2 = fminf(fmaxf(qcx + 0.5f * qw, 0.0f), Ww);
                const float y2 = fminf(fmaxf(qcy + 0.5f * qh, 0.0f), Hh);
                const bool valid = (sc > SCORE_T) &&
                                   ((x2 - x1) >= MINSZ) && ((y2 - y1) >= MINSZ);
                const int m = n * CM1 + (c - 1);
                bx[m] = make_float4(x1, y1, x2, y2);
                lv[m] = valid ? sc : -INFINITY;
            }
        }
    }
}

// ---------------------------------------------------------------------------
// Kernel 2: greedy batched NMS, one workgroup (32 waves) per image.
// Argmax over all M live scores (L2 resident), then suppression only over the
// winning class (cross-class IoU is exactly 0 under the reference's offsets).
// ---------------------------------------------------------------------------
__global__ __launch_bounds__(1024)
void nms_kernel(const float4* __restrict__ boxes_ws,
                float*        __restrict__ live_ws,
                float*        __restrict__ out)
{
    const int img  = blockIdx.x;
    const int tid  = threadIdx.x;
    const int lane = tid & 31;
    const int wid  = tid >> 5;

    const float4* bx = boxes_ws + (size_t)img * Mq;
    float*        lv = live_ws  + (size_t)img * Mq;
    float* outB = out + img * (DETS * 4);
    float* outS = out + Bq * DETS * 4 + img * DETS;
    float* outL = out + Bq * DETS * 5 + img * DETS;

    // zero-fill outputs (harness poisons d_out)
    for (int i = tid; i < DETS * 4; i += 1024) outB[i] = 0.0f;
    for (int i = tid; i < DETS;     i += 1024) { outS[i] = 0.0f; outL[i] = -1.0f; }

    __shared__ float swv[32];
    __shared__ int   swi[32];
    __shared__ float bb[5];
    __shared__ int   bsel;
    __syncthreads();

    for (int det = 0; det < DETS; ++det) {
        // ---- global argmax (first-max-wins like jnp.argmax) ----
        float bv = -INFINITY;
        int   bi = 0;
        for (int j = tid; j < Mq; j += 1024) {
            const float v = lv[j];
            if (v > bv || (v == bv && j < bi)) { bv = v; bi = j; }
        }
#pragma unroll
        for (int off = 16; off > 0; off >>= 1) {
            const float ov = __shfl_down(bv, off);
            const int   oi = __shfl_down(bi, off);
            if (ov > bv || (ov == bv && oi < bi)) { bv = ov; bi = oi; }
        }
        if (lane == 0) { swv[wid] = bv; swi[wid] = bi; }
        __syncthreads();
        if (wid == 0) {
            bv = swv[lane]; bi = swi[lane];
#pragma unroll
            for (int off = 16; off > 0; off >>= 1) {
                const float ov = __shfl_down(bv, off);
                const int   oi = __shfl_down(bi, off);
                if (ov > bv || (ov == bv && oi < bi)) { bv = ov; bi = oi; }
            }
            if (lane == 0) {
                if (bv == -INFINITY) {
                    bsel = -1;
                } else {
                    bsel = bi;
                    const float4 q = bx[bi];
                    bb[0] = q.x; bb[1] = q.y; bb[2] = q.z; bb[3] = q.w; bb[4] = bv;
                }
            }
        }
        __syncthreads();

        const int best = bsel;
        if (best < 0) break;   // uniform: everything left is -inf

        const float x1 = bb[0], y1 = bb[1], x2 = bb[2], y2 = bb[3];
        const int cls = best % CM1;          // label - 1

        if (tid == 0) {
            outB[det * 4 + 0] = x1; outB[det * 4 + 1] = y1;
            outB[det * 4 + 2] = x2; outB[det * 4 + 3] = y2;
            outS[det] = bb[4];
            outL[det] = (float)(cls + 1);
            lv[best] = -INFINITY;
        }

        // ---- suppression: only same-class candidates can overlap ----
        const float a1 = (x2 - x1) * (y2 - y1);
        for (int k = tid; k < Nq; k += 1024) {
            const int j = cls + k * CM1;
            const float4 q = bx[j];
            const float lx = fmaxf(x1, q.x), ly = fmaxf(y1, q.y);
            const float rx = fminf(x2, q.z), ry = fminf(y2, q.w);
            const float iw = fmaxf(rx - lx, 0.0f);
            const float ih = fmaxf(ry - ly, 0.0f);
            const float inter = iw * ih;
            const float a2 = (q.z - q.x) * (q.w - q.y);
            const float iou = inter / (a1 + a2 - inter);
            if (iou > NMS_T) lv[j] = -INFINITY;
        }
        __syncthreads();
    }
}

// ---------------------------------------------------------------------------
extern "C" void kernel_launch(void* const* d_in, const int* in_sizes, int n_in,
                              void* d_out, int out_size, void* d_ws, size_t ws_size,
                              hipStream_t stream)
{
    (void)in_sizes; (void)n_in; (void)out_size; (void)ws_size;
    const float* logits = (const float*)d_in[0];   // [B*N, C]
    const float* deltas = (const float*)d_in[1];   // [B*N, C*4]
    const float* props  = (const float*)d_in[2];   // [B, N, 4]
    const int*   shapes = (const int*)d_in[3];     // [B, 2] (H, W)
    float* out = (float*)d_out;

    // scratch layout: boxes [B*M] float4, then live [B*M] float (~29.5 MB)
    float4* boxes_ws = (float4*)d_ws;
    float*  live_ws  = (float*)((char*)d_ws + (size_t)Bq * Mq * sizeof(float4));

    // prep: one wave per 16 rows -> (B*N/16) waves = 1024 waves = 32768 threads
    const int threads = (Bq * Nq / 16) * 32;
    prep_kernel<<<threads / 256, 256, 0, stream>>>(logits, deltas, props, shapes,
                                                   boxes_ws, live_ws);
    // NMS: one 32-wave workgroup per image
    nms_kernel<<<Bq, 1024, 0, stream>>>(boxes_ws, live_ws, out);
}